// GCN_2078764171905
// MI455X (gfx1250) — compile-verified
//
#include <hip/hip_runtime.h>

// ---------------------------------------------------------------------------
// Problem constants (match reference)
// ---------------------------------------------------------------------------
static constexpr int Bc   = 16;
static constexpr int Mc   = 16;
static constexpr int Pc   = 128;
static constexpr int Dc   = 128;
static constexpr int Hc   = 8;
static constexpr int Lc   = 2;
static constexpr int DFc  = 256;
static constexpr int Nn   = Bc * Pc;          // 2048 graph nodes
static constexpr int Ec   = 32768;            // edges
static constexpr int Rr   = Bc * Mc * Pc;     // 32768 transformer rows

typedef float v2f __attribute__((ext_vector_type(2)));
typedef float v8f __attribute__((ext_vector_type(8)));

__device__ __forceinline__ v8f wmma4(v2f a, v2f b, v8f c) {
  // D = A(16x4,f32) * B(4x16,f32) + C(16x16,f32)
  return __builtin_amdgcn_wmma_f32_16x16x4_f32(
      /*neg_a=*/false, a, /*neg_b=*/false, b,
      /*c_mod=*/(short)0, c, /*reuse_a=*/false, /*reuse_b=*/false);
}

// ---------------------------------------------------------------------------
// Small utility kernels
// ---------------------------------------------------------------------------
__global__ void k_fill(float* __restrict__ p, float v, int n) {
  int i = blockIdx.x * blockDim.x + threadIdx.x;
  if (i < n) p[i] = v;
}

__global__ void k_deg_edges(const int* __restrict__ ei, float* __restrict__ deg) {
  int e = blockIdx.x * blockDim.x + threadIdx.x;
  if (e < Ec) atomicAdd(&deg[ei[Ec + e]], 1.0f);
}

__global__ void k_dinv(const float* __restrict__ deg, float* __restrict__ dinv) {
  int i = blockIdx.x * blockDim.x + threadIdx.x;
  if (i < Nn) {
    float d = deg[i];
    dinv[i] = d > 0.0f ? rsqrtf(d) : 0.0f;
  }
}

// agg[dst] += h[src] * dinv[src]*dinv[dst]  over all edges, per feature
__global__ void k_scatter(const int* __restrict__ ei, const float* __restrict__ h,
                          const float* __restrict__ dinv, float* __restrict__ agg) {
  int gid = blockIdx.x * blockDim.x + threadIdx.x;
  if (gid >= Ec * Dc) return;
  int e = gid >> 7;
  int f = gid & 127;
  int s = ei[e];
  int d = ei[Ec + e];
  float nrm = dinv[s] * dinv[d];
  atomicAdd(&agg[(size_t)d * Dc + f], h[(size_t)s * Dc + f] * nrm);
}

// out = relu(agg + h*dinv^2 (self loop) + bias)
__global__ void k_self_bias_relu(const float* __restrict__ agg, const float* __restrict__ h,
                                 const float* __restrict__ dinv, const float* __restrict__ bias,
                                 float* __restrict__ out) {
  int gid = blockIdx.x * blockDim.x + threadIdx.x;
  if (gid >= Nn * Dc) return;
  int i = gid >> 7;
  int f = gid & 127;
  float di = dinv[i];
  float v = agg[gid] + h[gid] * di * di + bias[f];
  out[gid] = fmaxf(v, 0.0f);
}

// X[b,m,p,:] = x2[b,p,:] + enc_in[b,m,p,:]   (float4 vectorized)
__global__ void k_bcast_add(const float* __restrict__ x2, const float* __restrict__ enc,
                            float* __restrict__ X) {
  int i4 = blockIdx.x * blockDim.x + threadIdx.x;   // over (B*M*P*D)/4
  if (i4 >= (Bc * Mc * Pc * Dc) / 4) return;
  int d4 = i4 & 31;                 // D/4 = 32
  int p  = (i4 >> 5) & 127;
  int b  = i4 >> 16;                // (i4 >> (5+7+4))
  const float4 xe = *(const float4*)(enc + (size_t)i4 * 4);
  const float4 xg = *(const float4*)(x2 + ((size_t)(b * Pc + p) * Dc + d4 * 4));
  float4 o;
  o.x = xg.x + xe.x; o.y = xg.y + xe.y; o.z = xg.z + xe.z; o.w = xg.w + xe.w;
  *(float4*)(X + (size_t)i4 * 4) = o;
}

// ---------------------------------------------------------------------------
// Generic WMMA f32 GEMM:  C[M,N] = A[M,K] @ B[K,N] (+bias) (+residual) (relu?)
// Block = 256 threads (8 waves). Wave w computes tile (blockIdx.x row-strip,
// n = blockIdx.y*128 + w*16). K tiled by 64 through LDS; B staged transposed
// so both fragments are single float2 DS loads.
// Requires: M%16==0, N%128==0, K%64==0 (true for all calls here).
// ---------------------------------------------------------------------------
__global__ __launch_bounds__(256) void k_wmma_gemm(
    const float* __restrict__ A, const float* __restrict__ Bm,
    const float* __restrict__ bias, const float* __restrict__ resid,
    float* __restrict__ C, int M, int N, int K, int relu) {
  __shared__ float As[16 * 64];     // A strip, row-major [16][64]
  __shared__ float Bt[128 * 64];    // B chunk transposed [n_local][k_local]

  const int tid   = threadIdx.x;
  const int lane  = tid & 31;
  const int wave  = tid >> 5;
  const int m0    = blockIdx.x * 16;
  const int nbase = blockIdx.y * 128;

  v8f acc = {0.f, 0.f, 0.f, 0.f, 0.f, 0.f, 0.f, 0.f};

  const int l15 = lane & 15;
  const int hi2 = (lane >> 4) * 2;

  for (int kc = 0; kc < K; kc += 64) {
    // --- stage A strip: 16x64 = 256 float4, one per thread
    {
      int row = tid >> 4;
      int c4  = (tid & 15) * 4;
      float4 v = *(const float4*)(A + (size_t)(m0 + row) * K + kc + c4);
      *(float4*)(&As[row * 64 + c4]) = v;
    }
    // --- stage B chunk transposed: 64x128 floats
    for (int i = 0; i < 8; ++i) {
      int idx = tid + i * 256;
      int row = idx >> 5;          // k_local 0..63
      int cg  = (idx & 31) * 4;    // n_local group
      float4 v = *(const float4*)(Bm + (size_t)(kc + row) * N + nbase + cg);
      Bt[(cg + 0) * 64 + row] = v.x;
      Bt[(cg + 1) * 64 + row] = v.y;
      Bt[(cg + 2) * 64 + row] = v.z;
      Bt[(cg + 3) * 64 + row] = v.w;
    }
    __syncthreads();

    // --- 16 WMMA k-steps over this chunk
    const float* bbase = &Bt[(wave * 16 + l15) * 64];
    const float* abase = &As[l15 * 64];
    for (int ks = 0; ks < 64; ks += 4) {
      v2f a = *(const v2f*)(abase + ks + hi2);
      v2f b = *(const v2f*)(bbase + ks + hi2);
      acc = wmma4(a, b, acc);
    }
    __syncthreads();
  }

  // --- epilogue: C 16x16 layout, VGPR i -> row (i | i+8), lane&15 -> col
  const int n = nbase + wave * 16 + l15;
  float bv = bias ? bias[n] : 0.0f;
#pragma unroll
  for (int i = 0; i < 8; ++i) {
    int m = m0 + i + (lane >> 4) * 8;
    float v = acc[i] + bv;
    if (resid) v += resid[(size_t)m * N + n];
    if (relu)  v = fmaxf(v, 0.0f);
    C[(size_t)m * N + n] = v;
  }
}

// ---------------------------------------------------------------------------
// Attention: one wave per (m-tile, head, batch). dh = 16, S = 128.
// scores = q@k^T * 0.25 -> softmax -> o = p@v. Writes o into O (may alias Q
// region this block staged, safe: staged to LDS before writes).
// ---------------------------------------------------------------------------
__global__ __launch_bounds__(32) void k_attention(
    const float* __restrict__ Q, const float* __restrict__ Kb,
    const float* __restrict__ V, float* __restrict__ O) {
  __shared__ float qs[16 * 16];     // q strip  [m][kk]
  __shared__ float ks_[128 * 16];   // K rows   [key][kk]
  __shared__ float vt[16 * 128];    // V^T      [dh][key]
  __shared__ float ps[16 * 128];    // probs    [m][key]

  const int lane = threadIdx.x;
  const int mt = blockIdx.x, h = blockIdx.y, bm = blockIdx.z;
  const size_t rowbase = (size_t)bm * 128;
  const int col0 = h * 16;

  // stage q (16x16)
#pragma unroll
  for (int i = 0; i < 2; ++i) {
    int idx = lane + i * 32;
    int r = idx >> 2, cg = (idx & 3) * 4;
    float4 v = *(const float4*)(Q + (rowbase + mt * 16 + r) * Dc + col0 + cg);
    *(float4*)(&qs[r * 16 + cg]) = v;
  }
  // stage K (128x16) and V^T (16x128)
#pragma unroll
  for (int i = 0; i < 16; ++i) {
    int idx = lane + i * 32;
    int r = idx >> 2, cg = (idx & 3) * 4;
    float4 kv = *(const float4*)(Kb + (rowbase + r) * Dc + col0 + cg);
    *(float4*)(&ks_[r * 16 + cg]) = kv;
    float4 vv = *(const float4*)(V + (rowbase + r) * Dc + col0 + cg);
    vt[(cg + 0) * 128 + r] = vv.x;
    vt[(cg + 1) * 128 + r] = vv.y;
    vt[(cg + 2) * 128 + r] = vv.z;
    vt[(cg + 3) * 128 + r] = vv.w;
  }
  __syncthreads();

  const int l15 = lane & 15;
  const int hi2 = (lane >> 4) * 2;

  // scores: 8 column tiles, K-dim = dh = 16 (4 wmma steps each)
  v8f sacc[8];
#pragma unroll
  for (int t = 0; t < 8; ++t) sacc[t] = (v8f){0.f,0.f,0.f,0.f,0.f,0.f,0.f,0.f};
#pragma unroll
  for (int t = 0; t < 8; ++t) {
    const float* bb = &ks_[(t * 16 + l15) * 16];
#pragma unroll
    for (int kk = 0; kk < 16; kk += 4) {
      v2f a = *(const v2f*)(&qs[l15 * 16 + kk + hi2]);
      v2f b = *(const v2f*)(bb + kk + hi2);
      sacc[t] = wmma4(a, b, sacc[t]);
    }
  }

  // softmax over rows (row = VGPR index; spread over 16-lane group * 8 tiles)
  const float scale = 0.25f;  // 1/sqrt(dh=16)
#pragma unroll
  for (int i = 0; i < 8; ++i) {
    float mx = -3.0e38f;
#pragma unroll
    for (int t = 0; t < 8; ++t) mx = fmaxf(mx, sacc[t][i] * scale);
    for (int d = 1; d < 16; d <<= 1) mx = fmaxf(mx, __shfl_xor(mx, d, 32));
    float sum = 0.0f;
#pragma unroll
    for (int t = 0; t < 8; ++t) {
      float e = __expf(sacc[t][i] * scale - mx);
      sacc[t][i] = e;
      sum += e;
    }
    for (int d = 1; d < 16; d <<= 1) sum += __shfl_xor(sum, d, 32);
    float inv = 1.0f / sum;
    int m = i + (lane >> 4) * 8;
#pragma unroll
    for (int t = 0; t < 8; ++t) ps[m * 128 + t * 16 + l15] = sacc[t][i] * inv;
  }
  __syncthreads();

  // o = p(16x128) @ v(128x16): 32 wmma k-steps
  v8f oacc = {0.f,0.f,0.f,0.f,0.f,0.f,0.f,0.f};
#pragma unroll
  for (int ks = 0; ks < 128; ks += 4) {
    v2f a = *(const v2f*)(&ps[l15 * 128 + ks + hi2]);
    v2f b = *(const v2f*)(&vt[l15 * 128 + ks + hi2]);
    oacc = wmma4(a, b, oacc);
  }

#pragma unroll
  for (int i = 0; i < 8; ++i) {
    int m = i + (lane >> 4) * 8;
    O[(rowbase + mt * 16 + m) * Dc + col0 + l15] = oacc[i];
  }
}

// ---------------------------------------------------------------------------
// LayerNorm over rows of 128. One wave per row, 8 rows per block.
// ---------------------------------------------------------------------------
__global__ __launch_bounds__(256) void k_layernorm(
    const float* __restrict__ X, const float* __restrict__ g,
    const float* __restrict__ b, float* __restrict__ Y, int rows) {
  int wave = threadIdx.x >> 5;
  int lane = threadIdx.x & 31;
  int row = blockIdx.x * 8 + wave;
  if (row >= rows) return;
  float4 v = *(const float4*)(X + (size_t)row * Dc + lane * 4);
  float s  = v.x + v.y + v.z + v.w;
  float ss = v.x * v.x + v.y * v.y + v.z * v.z + v.w * v.w;
  for (int d = 1; d < 32; d <<= 1) {
    s  += __shfl_xor(s, d, 32);
    ss += __shfl_xor(ss, d, 32);
  }
  float mu  = s * (1.0f / 128.0f);
  float var = ss * (1.0f / 128.0f) - mu * mu;
  float r   = rsqrtf(var + 1e-5f);
  float4 gv = *(const float4*)(g + lane * 4);
  float4 bv = *(const float4*)(b + lane * 4);
  float4 o;
  o.x = (v.x - mu) * r * gv.x + bv.x;
  o.y = (v.y - mu) * r * gv.y + bv.y;
  o.z = (v.z - mu) * r * gv.z + bv.z;
  o.w = (v.w - mu) * r * gv.w + bv.w;
  *(float4*)(Y + (size_t)row * Dc + lane * 4) = o;
}

// ---------------------------------------------------------------------------
// Orchestration
// ---------------------------------------------------------------------------
extern "C" void kernel_launch(void* const* d_in, const int* in_sizes, int n_in,
                              void* d_out, int out_size, void* d_ws, size_t ws_size,
                              hipStream_t stream) {
  (void)in_sizes; (void)n_in; (void)out_size; (void)ws_size;

  const float* enc_in = (const float*)d_in[1];
  const float* x_raw  = (const float*)d_in[2];
  const int*   ei     = (const int*)d_in[3];
  const float* gw1 = (const float*)d_in[4];
  const float* gb1 = (const float*)d_in[5];
  const float* gw2 = (const float*)d_in[6];
  const float* gb2 = (const float*)d_in[7];
  const float* wq = (const float*)d_in[8];
  const float* wk = (const float*)d_in[9];
  const float* wv = (const float*)d_in[10];
  const float* wo = (const float*)d_in[11];
  const float* bq = (const float*)d_in[12];
  const float* bk = (const float*)d_in[13];
  const float* bv = (const float*)d_in[14];
  const float* bo = (const float*)d_in[15];
  const float* fw1 = (const float*)d_in[16];
  const float* fb1 = (const float*)d_in[17];
  const float* fw2 = (const float*)d_in[18];
  const float* fb2 = (const float*)d_in[19];
  const float* ln1g = (const float*)d_in[20];
  const float* ln1b = (const float*)d_in[21];
  const float* ln2g = (const float*)d_in[22];
  const float* ln2b = (const float*)d_in[23];

  float* W = (float*)d_ws;
  const size_t RD = (size_t)Rr * Dc;         // 4,194,304 floats
  float* X   = W;                            // [R,128]
  float* Qb  = W + RD;                       // [R,128] (also attention output)
  float* Kb  = W + 2 * RD;
  float* Vb  = W + 3 * RD;
  float* TMP = W + 4 * RD;                   // [R,128]
  float* FFH = W + 5 * RD;                   // [R,256]  (8,388,608 floats)
  // GCN scratch overlapped with FFH region (GCN finishes before FFN use):
  float* deg  = FFH;                         // [N]
  float* dinv = FFH + Nn;                    // [N]
  float* h    = FFH + 2 * Nn;                // [N,128]
  float* agg  = h + (size_t)Nn * Dc;
  float* x1   = agg + (size_t)Nn * Dc;
  float* x2   = x1 + (size_t)Nn * Dc;

  const int T = 256;
  const int ND = Nn * Dc;

  // ---- GCN layer 1 ----
  k_fill<<<(Nn + T - 1) / T, T, 0, stream>>>(deg, 1.0f, Nn);   // self loops
  k_fill<<<(ND + T - 1) / T, T, 0, stream>>>(agg, 0.0f, ND);
  k_deg_edges<<<Ec / T, T, 0, stream>>>(ei, deg);
  k_dinv<<<(Nn + T - 1) / T, T, 0, stream>>>(deg, dinv);
  k_wmma_gemm<<<dim3(Nn / 16, 1), T, 0, stream>>>(x_raw, gw1, nullptr, nullptr, h, Nn, Dc, Dc, 0);
  k_scatter<<<(Ec * Dc) / T, T, 0, stream>>>(ei, h, dinv, agg);
  k_self_bias_relu<<<ND / T, T, 0, stream>>>(agg, h, dinv, gb1, x1);

  // ---- GCN layer 2 ----
  k_fill<<<(ND + T - 1) / T, T, 0, stream>>>(agg, 0.0f, ND);
  k_wmma_gemm<<<dim3(Nn / 16, 1), T, 0, stream>>>(x1, gw2, nullptr, nullptr, h, Nn, Dc, Dc, 0);
  k_scatter<<<(Ec * Dc) / T, T, 0, stream>>>(ei, h, dinv, agg);
  k_self_bias_relu<<<ND / T, T, 0, stream>>>(agg, h, dinv, gb2, x2);

  // ---- broadcast over M + enc_in ----
  k_bcast_add<<<(Bc * Mc * Pc * Dc / 4) / T, T, 0, stream>>>(x2, enc_in, X);

  // ---- transformer layers ----
  for (int l = 0; l < Lc; ++l) {
    const float* wql = wq + (size_t)l * Dc * Dc;
    const float* wkl = wk + (size_t)l * Dc * Dc;
    const float* wvl = wv + (size_t)l * Dc * Dc;
    const float* wol = wo + (size_t)l * Dc * Dc;
    const float* bql = bq + (size_t)l * Dc;
    const float* bkl = bk + (size_t)l * Dc;
    const float* bvl = bv + (size_t)l * Dc;
    const float* bol = bo + (size_t)l * Dc;
    const float* f1l = fw1 + (size_t)l * Dc * DFc;
    const float* g1l = fb1 + (size_t)l * DFc;
    const float* f2l = fw2 + (size_t)l * DFc * Dc;
    const float* g2l = fb2 + (size_t)l * Dc;

    dim3 gD(Rr / 16, 1);
    k_wmma_gemm<<<gD, T, 0, stream>>>(X, wql, bql, nullptr, Qb, Rr, Dc, Dc, 0);
    k_wmma_gemm<<<gD, T, 0, stream>>>(X, wkl, bkl, nullptr, Kb, Rr, Dc, Dc, 0);
    k_wmma_gemm<<<gD, T, 0, stream>>>(X, wvl, bvl, nullptr, Vb, Rr, Dc, Dc, 0);

    // attention; output written over Qb (each block only reads its own region)
    k_attention<<<dim3(8, Hc, Bc * Mc), 32, 0, stream>>>(Qb, Kb, Vb, Qb);

    // o-proj + residual, then LN1 -> X
    k_wmma_gemm<<<gD, T, 0, stream>>>(Qb, wol, bol, X, TMP, Rr, Dc, Dc, 0);
    k_layernorm<<<Rr / 8, T, 0, stream>>>(TMP, ln1g + l * Dc, ln1b + l * Dc, X, Rr);

    // FFN: relu(X@fw1+fb1) -> FFH ; FFH@fw2+fb2 + X -> TMP ; LN2
    k_wmma_gemm<<<dim3(Rr / 16, 2), T, 0, stream>>>(X, f1l, g1l, nullptr, FFH, Rr, DFc, Dc, 1);
    k_wmma_gemm<<<gD, T, 0, stream>>>(FFH, f2l, g2l, X, TMP, Rr, Dc, DFc, 0);
    float* dst = (l == Lc - 1) ? (float*)d_out : X;
    k_layernorm<<<Rr / 8, T, 0, stream>>>(TMP, ln2g + l * Dc, ln2b + l * Dc, dst, Rr);
  }
}